// Agent_84524956385608
// MI455X (gfx1250) — compile-verified
//
#include <hip/hip_runtime.h>
#include <hip/hip_bf16.h>

// Problem constants (match reference)
#define NN 200000
#define EE 3200000
#define GG 1024

typedef __attribute__((ext_vector_type(2))) float v2f;
typedef __attribute__((ext_vector_type(8))) float v8f;

// ---------------- degree / normalization ----------------
__global__ void k_fill(float* p, float v, int n) {
    int i = blockIdx.x * blockDim.x + threadIdx.x;
    if (i < n) p[i] = v;
}

__global__ void k_deg_edges(const int* __restrict__ dst, float* __restrict__ deg, int e) {
    int i = blockIdx.x * blockDim.x + threadIdx.x;
    if (i < e) atomicAdd(&deg[dst[i]], 1.0f);
}

__global__ void k_rsqrt_inplace(float* __restrict__ p, int n) {
    int i = blockIdx.x * blockDim.x + threadIdx.x;
    if (i < n) p[i] = rsqrtf(p[i]);   // deg >= 1 always (self-loop)
}

// ---------------- layer-1 aggregation (8 features) ----------------
__global__ void k_agg8_init(const float* __restrict__ x, const float* __restrict__ dinv,
                            float* __restrict__ agg, int n8) {
    int i = blockIdx.x * blockDim.x + threadIdx.x;
    if (i < n8) {
        float di = dinv[i >> 3];
        agg[i] = x[i] * di * di;      // self-loop term, norm = dinv[v]^2
    }
}

__global__ void k_agg8_edges(const int* __restrict__ src, const int* __restrict__ dst,
                             const float* __restrict__ x, const float* __restrict__ dinv,
                             float* __restrict__ agg, int e8) {
    int t = blockIdx.x * blockDim.x + threadIdx.x;
    if (t >= e8) return;
    int e = t >> 3, f = t & 7;
    int s = src[e], d = dst[e];
    float c = dinv[s] * dinv[d];
    atomicAdd(&agg[(size_t)d * 8 + f], x[(size_t)s * 8 + f] * c);
}

// ---------------- layer-2 aggregation (128 features, wave per edge) ----------------
__global__ void k_agg128_init(const float* __restrict__ h, const float* __restrict__ dinv,
                              float* __restrict__ agg, int n32) {
    int t = blockIdx.x * blockDim.x + threadIdx.x;
    if (t >= n32) return;
    int v = t >> 5, f4 = (t & 31) * 4;
    float di = dinv[v]; float s = di * di;
    const float4 hv = *(const float4*)(h + (size_t)v * 128 + f4);
    float4 o; o.x = hv.x * s; o.y = hv.y * s; o.z = hv.z * s; o.w = hv.w * s;
    *(float4*)(agg + (size_t)v * 128 + f4) = o;
}

__global__ void k_agg128_edges(const int* __restrict__ src, const int* __restrict__ dst,
                               const float* __restrict__ h, const float* __restrict__ dinv,
                               float* __restrict__ agg, int e32) {
    int t = blockIdx.x * blockDim.x + threadIdx.x;
    if (t >= e32) return;
    int e = t >> 5, f4 = (t & 31) * 4;           // whole wave shares one edge
    int s = src[e], d = dst[e];                   // broadcast loads
    float c = dinv[s] * dinv[d];
    const float4 hv = *(const float4*)(h + (size_t)s * 128 + f4);  // coalesced 512B
    float* out = agg + (size_t)d * 128 + f4;
    atomicAdd(out + 0, hv.x * c);
    atomicAdd(out + 1, hv.y * c);
    atomicAdd(out + 2, hv.z * c);
    atomicAdd(out + 3, hv.w * c);
}

// ---------------- generic fp32 WMMA GEMM: out = relu(A[M,K] @ W[K,Nout] + b) ----------------
// One 16x16 tile per wave, V_WMMA_F32_16X16X4_F32 over the K loop (exact fp32 FMA).
// If batch != nullptr: fused global-max-pool -> atomicMax(pooled[batch[m]*Nout + n])
// (values are >= 0 post-ReLU so int-bit atomicMax on 0-initialized pooled is exact).
__global__ void k_wmma_gemm(const float* __restrict__ A, const float* __restrict__ W,
                            const float* __restrict__ bias, float* __restrict__ out,
                            const int* __restrict__ batch, int M, int K, int Nout) {
    int lane = threadIdx.x & 31;
    int wave = threadIdx.x >> 5;
    int tile = blockIdx.x * 4 + wave;            // grid sized exactly: tiles % 4 == 0
    int tilesN = Nout >> 4;
    int tileM = tile / tilesN;
    int tileN = tile - tileM * tilesN;

    int half = lane >> 4;                        // 0 or 1
    int mrow = lane & 15;                        // A row within tile (both halves)
    int col  = lane & 15;                        // B/C/D column within tile
    int koff = half * 2;                         // ISA 7.12.2: VGPR0 holds K=0 / K=2

    const float* Arow = A + (size_t)(tileM * 16 + mrow) * K;
    const float* Wc   = W + (size_t)(tileN * 16 + col);

    v8f acc = {};
    for (int kk = 0; kk < K; kk += 4) {
        int k0 = kk + koff;
        v2f a; a.x = Arow[k0];                 // A[m][k0], A[m][k0+1]
               a.y = Arow[k0 + 1];
        v2f b; b.x = Wc[(size_t)k0 * Nout];    // B[k0][n], B[k0+1][n]
               b.y = Wc[(size_t)(k0 + 1) * Nout];
        acc = __builtin_amdgcn_wmma_f32_16x16x4_f32(
                  false, a, false, b, (short)0, acc, false, false);
    }

    int n = tileN * 16 + col;
    float bv = bias[n];
    if (batch) {
        for (int r = 0; r < 8; ++r) {
            int m = tileM * 16 + r + 8 * half;   // C/D layout: VGPR r -> row r (+8 upper half)
            float v = acc[r] + bv;
            v = v > 0.0f ? v : 0.0f;
            int g = batch[m];
            atomicMax((int*)&out[(size_t)g * Nout + n], __float_as_int(v));
        }
    } else {
        for (int r = 0; r < 8; ++r) {
            int m = tileM * 16 + r + 8 * half;
            float v = acc[r] + bv;
            out[(size_t)m * Nout + n] = v > 0.0f ? v : 0.0f;
        }
    }
}

// ---------------- host-side launch ----------------
static inline int cdiv(int a, int b) { return (a + b - 1) / b; }

extern "C" void kernel_launch(void* const* d_in, const int* in_sizes, int n_in,
                              void* d_out, int out_size, void* d_ws, size_t ws_size,
                              hipStream_t stream) {
    const float* x     = (const float*)d_in[0];
    const int*   ei    = (const int*)d_in[1];
    const int*   src   = ei;
    const int*   dst   = ei + EE;
    const int*   batch = (const int*)d_in[2];
    const float* W1    = (const float*)d_in[3];
    const float* b1    = (const float*)d_in[4];
    const float* W2    = (const float*)d_in[5];
    const float* b2    = (const float*)d_in[6];
    const float* fcW1  = (const float*)d_in[7];
    const float* fcb1  = (const float*)d_in[8];
    const float* fcW2  = (const float*)d_in[9];
    const float* fcb2  = (const float*)d_in[10];
    const float* fcW3  = (const float*)d_in[11];
    const float* fcb3  = (const float*)d_in[12];
    const float* fcW4  = (const float*)d_in[13];
    const float* fcb4  = (const float*)d_in[14];

    // workspace layout (floats)
    float* ws    = (float*)d_ws;
    float* dinv  = ws;                                  // N          (deg computed in-place)
    float* aggx  = dinv + NN;                           // N*8
    float* h1    = aggx + (size_t)NN * 8;               // N*128
    float* agg2  = h1   + (size_t)NN * 128;             // N*128
    float* pooled= agg2 + (size_t)NN * 128;             // G*128
    float* f1    = pooled + (size_t)GG * 128;           // G*128
    float* f2    = f1   + (size_t)GG * 128;             // G*64
    float* f3    = f2   + (size_t)GG * 64;              // G*32

    const int B = 256;

    // 1) symmetric normalization: deg = indeg + 1 ; dinv = rsqrt(deg)
    k_fill<<<cdiv(NN, B), B, 0, stream>>>(dinv, 1.0f, NN);
    k_deg_edges<<<cdiv(EE, B), B, 0, stream>>>(dst, dinv, EE);
    k_rsqrt_inplace<<<cdiv(NN, B), B, 0, stream>>>(dinv, NN);

    // 2) layer-1: aggregate x first (8 feats), then GEMM (linearity of GCN)
    k_agg8_init<<<cdiv(NN * 8, B), B, 0, stream>>>(x, dinv, aggx, NN * 8);
    k_agg8_edges<<<cdiv(EE * 8, B), B, 0, stream>>>(src, dst, x, dinv, aggx, EE * 8);
    // h1 = relu(aggx @ W1 + b1)   [200000 x 8 x 128] -> 100000 tiles / 4 waves
    k_wmma_gemm<<<(NN / 16) * (128 / 16) / 4, 128, 0, stream>>>(
        aggx, W1, b1, h1, nullptr, NN, 8, 128);

    // 3) layer-2: aggregate h1 (128 feats), GEMM fused with bias+relu+segment-max-pool
    k_agg128_init<<<cdiv(NN * 32, B), B, 0, stream>>>(h1, dinv, agg2, NN * 32);
    k_agg128_edges<<<cdiv(EE * 32, B), B, 0, stream>>>(src, dst, h1, dinv, agg2, EE * 32);
    k_fill<<<cdiv(GG * 128, B), B, 0, stream>>>(pooled, 0.0f, GG * 128);
    k_wmma_gemm<<<(NN / 16) * (128 / 16) / 4, 128, 0, stream>>>(
        agg2, W2, b2, pooled, batch, NN, 128, 128);

    // 4) FC head on pooled [1024,128]
    k_wmma_gemm<<<(GG / 16) * (128 / 16) / 4, 128, 0, stream>>>(
        pooled, fcW1, fcb1, f1, nullptr, GG, 128, 128);
    k_wmma_gemm<<<(GG / 16) * (64 / 16) / 4, 128, 0, stream>>>(
        f1, fcW2, fcb2, f2, nullptr, GG, 128, 64);
    k_wmma_gemm<<<(GG / 16) * (32 / 16) / 4, 128, 0, stream>>>(
        f2, fcW3, fcb3, f3, nullptr, GG, 64, 32);
    k_wmma_gemm<<<(GG / 16) * (16 / 16) / 4, 128, 0, stream>>>(
        f3, fcW4, fcb4, (float*)d_out, nullptr, GG, 32, 16);
}